// fUpModule_62534723830150
// MI455X (gfx1250) — compile-verified
//
#include <hip/hip_runtime.h>

typedef __attribute__((ext_vector_type(2))) float v2f;
typedef __attribute__((ext_vector_type(8))) float v8f;

#define NTRI 69
#define NL   6

struct Meta {
    int l1[NTRI]; int l2[NTRI]; int l[NTRI];
    int cg_off[NTRI];     // float offset of this triple's (a,c) coeff table in d_ws
    int ko[NTRI];         // channel offset (ts base) within this l's middle-tau axis
    int cg_total;
    int mid[NL];          // MIDDLE_TAUS[l]
    int so[NL];           // moving_std offset per l
    int wo[NL];           // weights complex offset per l
    int f_off[NL];        // activation complex channel offset of F[l]
    int oo[NL];           // output complex channel offset per l
    int ntri_l[NL];
    int tri_l[NL][16];
};

constexpr Meta make_meta() {
    Meta m{};
    int cnt = 0, cgoff = 0;
    int midc[NL] = {0, 0, 0, 0, 0, 0};
    for (int l1 = 0; l1 <= 5; ++l1)
        for (int l2 = 0; l2 <= l1; ++l2) {
            int lmax = (l1 + l2 < 5) ? (l1 + l2) : 5;
            for (int l = l1 - l2; l <= lmax; ++l) {
                m.l1[cnt] = l1; m.l2[cnt] = l2; m.l[cnt] = l;
                m.cg_off[cnt] = cgoff; cgoff += (2 * l1 + 1) * (2 * l2 + 1);
                m.ko[cnt] = midc[l]; midc[l] += 64;   // TAUS[l1]*TAUS[l2] = 64
                m.tri_l[l][m.ntri_l[l]] = cnt; m.ntri_l[l] += 1;
                ++cnt;
            }
        }
    m.cg_total = cgoff;
    int so = 0, wo = 0, fo = 0, oo = 0;
    for (int l = 0; l < NL; ++l) {
        m.mid[l] = midc[l];
        m.so[l] = so;  so += midc[l];
        m.wo[l] = wo;  wo += 8 * midc[l];
        m.f_off[l] = fo; fo += 8 * (2 * l + 1);
        m.oo[l] = oo;  oo += 8 * (2 * l + 1);
    }
    return m;
}

constexpr Meta MT = make_meta();

// ---------------------------------------------------------------------------
// Kernel 1: compute Clebsch-Gordan coefficient tables into d_ws.
// ---------------------------------------------------------------------------
__device__ __forceinline__ double dfact(int n) {
    double r = 1.0;
    for (int i = 2; i <= n; ++i) r *= (double)i;
    return r;
}

__global__ void cg_init_kernel(float* __restrict__ cg) {
    const int ti = blockIdx.x;
    const int l1 = MT.l1[ti], l2 = MT.l2[ti], l = MT.l[ti];
    const int n1 = 2 * l1 + 1, n2 = 2 * l2 + 1;
    for (int e = threadIdx.x; e < n1 * n2; e += blockDim.x) {
        const int a = e / n2, c = e % n2;
        const int m1 = a - l1, m2 = c - l2, m = m1 + m2;
        double v = 0.0;
        if (m >= -l && m <= l) {
            double pre = sqrt((double)(2 * l + 1) * dfact(l + l1 - l2) * dfact(l - l1 + l2) *
                              dfact(l1 + l2 - l) / dfact(l1 + l2 + l + 1));
            pre *= sqrt(dfact(l + m) * dfact(l - m) * dfact(l1 - m1) * dfact(l1 + m1) *
                        dfact(l2 - m2) * dfact(l2 + m2));
            double s = 0.0;
            for (int k = 0; k <= l1 + l2 - l; ++k) {
                const int d0 = k, d1 = l1 + l2 - l - k, d2 = l1 - m1 - k;
                const int d3 = l2 + m2 - k, d4 = l - l2 + m1 + k, d5 = l - l1 - m2 + k;
                if (d0 < 0 || d1 < 0 || d2 < 0 || d3 < 0 || d4 < 0 || d5 < 0) continue;
                const double den = dfact(d0) * dfact(d1) * dfact(d2) * dfact(d3) * dfact(d4) * dfact(d5);
                s += ((k & 1) ? -1.0 : 1.0) / den;
            }
            v = pre * s;
        }
        cg[MT.cg_off[ti] + e] = (float)v;
    }
}

// ---------------------------------------------------------------------------
// Kernel 2: fused CG tensor product + normalization + complex channel-mix GEMM.
// grid = (64 batch tiles, 6 l).  128 threads = 4 waves; wave w owns m = w, w+4, w+8.
// GEMM mapping per (triple, m): M=16 batches, N=(o,re/im)=16, K=(ts,re/im)=128
//   -> 32 x V_WMMA_F32_16X16X4_F32 accumulations.
// Slot (m) loop is OUTER, q-loop inner: slot conditions / m1 bounds / CG row
// pointers are once-per-triple scalars; inner loop is ds_load + FMA + WMMA with
// pure pointer-increment addressing.
// ---------------------------------------------------------------------------
__global__ __launch_bounds__(128) void cg_fused_kernel(const float* __restrict__ acts,
                                                       const float* __restrict__ wts,
                                                       const float* __restrict__ stdv,
                                                       const float* __restrict__ cg,
                                                       float* __restrict__ out) {
    const int bt   = blockIdx.x;   // batch tile of 16
    const int l    = blockIdx.y;   // output l
    const int tid  = threadIdx.x;
    const int lane = tid & 31;
    // Wave index as a compiler-provable scalar: uniform control flow for all
    // m-dependent branches/loops (keeps WMMA outside any EXEC-masked region).
    const int w    = __builtin_amdgcn_readfirstlane(tid >> 5);

    __shared__ float s_act[16 * 576];  // 16 batches x 288 complex channels  (36 KB)
    __shared__ float s_B[2048];        // 128x16 B matrix for current triple (8 KB)

    // Stage activations tile (contiguous) as float4.
    {
        const float4* g = (const float4*)(acts + (size_t)bt * 16 * 576);
        float4* s = (float4*)s_act;
        for (int i = tid; i < 16 * 576 / 4; i += 128) s[i] = g[i];
    }
    __syncthreads();

    const int nm   = 2 * l + 1;
    const int midl = MT.mid[l];
    v8f acc[3] = {};

    const int b = lane & 15;
    const float* actb = s_act + b * 576;
    const int hi16 = lane >> 4;        // 0 for lanes 0-15, 1 for lanes 16-31

    const int ntri = MT.ntri_l[l];
    for (int tt = 0; tt < ntri; ++tt) {
        const int ti = MT.tri_l[l][tt];
        const int L1 = MT.l1[ti], L2 = MT.l2[ti];
        const int n1 = 2 * L1 + 1, n2 = 2 * L2 + 1;
        const float* cgt = cg + MT.cg_off[ti];
        const int ko = MT.ko[ti];
        const float* stb = stdv + MT.so[l] + ko;
        const float* wb  = wts + 2 * (size_t)MT.wo[l];

        // Prefetch next triple's weight slab into cache while we compute.
        if (tt + 1 < ntri) {
            const int kon = MT.ko[MT.tri_l[l][tt + 1]];
            __builtin_prefetch(wb + 2 * (size_t)kon, 0, 1);
        }

        // ---- stage B[(ts,ri_in) x (o,ri_out)] * inv_std, in WMMA lane layout ----
        __syncthreads();  // previous iteration's B reads complete
        for (int e = tid; e < 2048; e += 128) {
            const int q  = e >> 6;
            const int Ln = (e >> 1) & 31;
            const int u  = e & 1;
            const int kap = 4 * q + ((Ln & 16) ? 2 : 0) + u;
            const int n   = Ln & 15;
            const int ts  = kap >> 1, ri = kap & 1;
            const int o   = n >> 1,  ro = n & 1;
            const float wr = wb[2 * ((size_t)o * midl + ko + ts) + 0];
            const float wi = wb[2 * ((size_t)o * midl + ko + ts) + 1];
            const float inv = 1.0f / (stb[ts] + 1e-5f);
            float v;
            if (ri == 0) v = (ro == 0) ?  wr : wi;   // real input column
            else         v = (ro == 0) ? -wi : wr;   // imag input column
            s_B[e] = v * inv;
        }
        __syncthreads();

        const float* f1b = actb + 2 * MT.f_off[L1];
        const float* f2b = actb + 2 * MT.f_off[L2];

#pragma unroll
        for (int slot = 0; slot < 3; ++slot) {
            const int m = w + 4 * slot;          // scalar
            if (m >= nm) break;                  // scalar branch, once per triple
            const int mc = m - l;                // scalar
            int lo = -L1; if (mc - L2 > lo) lo = mc - L2;   // scalar bounds
            int hi =  L1; if (mc + L2 < hi) hi = mc + L2;
            const int ncf  = hi - lo + 1;        // scalar trip count (>=1 always)
            const int a0   = lo + L1;            // starting column in F1
            const int c0   = (mc - lo) + L2;     // starting column in F2
            const float* cg0 = cgt + a0 * n2 + c0;  // scalar CG row pointer
            const int cstep = n2 - 1;            // scalar CG pointer stride

            v8f a_acc = acc[slot];
            const float* bp = s_B + lane * 2;    // B operand pointer, +64 floats/q
            for (int q = 0; q < 32; ++q) {
                const int ts = 2 * q + hi16;
                const int t  = ts >> 3, s2 = ts & 7;
                const float* p1 = f1b + 2 * (t  * n1 + a0);
                const float* p2 = f2b + 2 * (s2 * n2 + c0);
                const float* cp = cg0;
                float fr = 0.0f, fi = 0.0f;
                for (int j = 0; j < ncf; ++j) {
                    const float coef = *cp;          // scalar (constant-cache) load
                    const float x1r = p1[0], x1i = p1[1];
                    const float x2r = p2[0], x2i = p2[1];
                    cp += cstep;                     // scalar pointer increment
                    p1 += 2;                         // a ascending
                    p2 -= 2;                         // c descending
                    fr += coef * (x1r * x2r - x1i * x2i);
                    fi += coef * (x1r * x2i + x1i * x2r);
                }
                const v2f Av = {fr, fi};
                const v2f Bv = *(const v2f*)bp;
                bp += 64;
                a_acc = __builtin_amdgcn_wmma_f32_16x16x4_f32(
                    false, Av, false, Bv, (short)0, a_acc, false, false);
            }
            acc[slot] = a_acc;
        }
    }

    // ---- write back: D rows = batch, cols = (o, re/im) ----
#pragma unroll
    for (int slot = 0; slot < 3; ++slot) {
        const int m = w + 4 * slot;    // scalar
        if (m >= nm) break;            // scalar branch
        const int n  = lane & 15;
        const int o  = n >> 1, ro = n & 1;
        const int rb = (lane & 16) ? 8 : 0;
        const int ch = MT.oo[l] + o * nm + m;
#pragma unroll
        for (int j = 0; j < 8; ++j) {
            const int bM = bt * 16 + rb + j;
            out[((size_t)bM * 288 + ch) * 2 + ro] = acc[slot][j];
        }
    }
}

// ---------------------------------------------------------------------------
extern "C" void kernel_launch(void* const* d_in, const int* in_sizes, int n_in,
                              void* d_out, int out_size, void* d_ws, size_t ws_size,
                              hipStream_t stream) {
    const float* acts = (const float*)d_in[0];   // (1024, 288, 2) f32
    const float* wts  = (const float*)d_in[1];   // (35328, 2) f32
    const float* stdv = (const float*)d_in[2];   // (4416,) f32
    float* out = (float*)d_out;                  // (1024, 288, 2) f32
    float* cg  = (float*)d_ws;                   // CG coefficient tables

    cg_init_kernel<<<dim3(NTRI), dim3(128), 0, stream>>>(cg);
    cg_fused_kernel<<<dim3(64, 6), dim3(128), 0, stream>>>(acts, wts, stdv, cg, out);
}